// SpectralCrossAttention_67250597921252
// MI455X (gfx1250) — compile-verified
//
#include <hip/hip_runtime.h>
#include <stdint.h>

#define HH 1024
#define WW 1024
#define NPIX (HH*WW)
#define BIGL NPIX
#define KSEG 16

// seg-sums geometry: 512 blocks x 256 threads = 4096 waves; each wave owns a
// contiguous 256-pixel range => exactly 64 chunks of 4 pixels (compile-time).
#define SEG_BLOCKS 512
#define SEG_TPB    256
#define SEG_WAVES  (SEG_BLOCKS * SEG_TPB / 32)
#define SEG_CHUNKS (NPIX / (4 * SEG_WAVES))   // = 64

typedef float v2f __attribute__((ext_vector_type(2)));
typedef float v8f __attribute__((ext_vector_type(8)));

// ---------------------------------------------------------------------------
// 1) Initial labels + aux plane for B: lab = mask ? pixel_index : BIG
//    auxB[q] = mean(img_B[q, :])  (precomputed so seg-sums has no 3-ch loads)
// ---------------------------------------------------------------------------
__global__ void k_init_labels(const float* __restrict__ gateA,
                              const unsigned char* __restrict__ contA,
                              const float* __restrict__ fiedB,
                              const float* __restrict__ imgB,
                              int* __restrict__ labA, int* __restrict__ labB,
                              float* __restrict__ auxB) {
  int p = blockIdx.x * blockDim.x + threadIdx.x;
  if (p < NPIX) {
    bool mA = (gateA[p] < 0.5f) && (contA[p] == 0);
    labA[p] = mA ? p : BIGL;
  } else if (p < 2 * NPIX) {
    int q = p - NPIX;
    labB[q] = (fiedB[q] > 0.0f) ? q : BIGL;
    auxB[q] = (imgB[3 * q] + imgB[3 * q + 1] + imgB[3 * q + 2]) * (1.0f / 3.0f);
  }
}

// ---------------------------------------------------------------------------
// 2) One CC min-propagation step over 4-neighborhood (both images, ping-pong)
//    Non-mask pixels carry BIG; mask[p] <=> cur[p] < BIG (labels only shrink).
// ---------------------------------------------------------------------------
__device__ __forceinline__ int cc_min(const int* __restrict__ cur, int p) {
  int l = cur[p];
  if (l >= BIGL) return BIGL;
  int y = p >> 10, x = p & (WW - 1);
  int up = (y > 0)      ? cur[p - WW] : BIGL;
  int dn = (y < HH - 1) ? cur[p + WW] : BIGL;
  int lf = (x > 0)      ? cur[p - 1]  : BIGL;
  int rt = (x < WW - 1) ? cur[p + 1]  : BIGL;
  int m = min(min(up, dn), min(lf, rt));
  return min(l, m);
}

__global__ void k_cc_step(const int* __restrict__ curA, int* __restrict__ nxtA,
                          const int* __restrict__ curB, int* __restrict__ nxtB) {
  int p = blockIdx.x * blockDim.x + threadIdx.x;
  if (p < NPIX) {
    // prefetch the next row (speculative; emits global_prefetch_b8)
    __builtin_prefetch(curA + ((p + WW < NPIX) ? p + WW : p), 0, 3);
    nxtA[p] = cc_min(curA, p);
  } else if (p < 2 * NPIX) {
    int q = p - NPIX;
    __builtin_prefetch(curB + ((q + WW < NPIX) ? q + WW : q), 0, 3);
    nxtB[q] = cc_min(curB, q);
  }
}

// ---------------------------------------------------------------------------
// 3) Bincount of labels (both images)
// ---------------------------------------------------------------------------
__global__ void k_bincount(const int* __restrict__ labA, const int* __restrict__ labB,
                           int* __restrict__ cntA, int* __restrict__ cntB) {
  int p = blockIdx.x * blockDim.x + threadIdx.x;
  if (p < NPIX) {
    int l = labA[p];
    if (l < BIGL) atomicAdd(&cntA[l], 1);
  } else if (p < 2 * NPIX) {
    int l = labB[p - NPIX];
    if (l < BIGL) atomicAdd(&cntB[l], 1);
  }
}

// ---------------------------------------------------------------------------
// 4) One argmax pass: key = (count<<32) | (0xFFFFFFFF - label)
//    (max key => largest count, ties -> smallest label, matching lax.top_k)
// ---------------------------------------------------------------------------
__global__ void k_argmax(const int* __restrict__ cntA, const int* __restrict__ cntB,
                         unsigned long long* __restrict__ bestA,
                         unsigned long long* __restrict__ bestB) {
  __shared__ unsigned long long sh[256];
  unsigned long long kA = 0ull, kB = 0ull;
  for (int i = blockIdx.x * blockDim.x + threadIdx.x; i < NPIX;
       i += gridDim.x * blockDim.x) {
    unsigned long long lo = (unsigned long long)(0xFFFFFFFFu - (unsigned)i);
    unsigned long long a = ((unsigned long long)(unsigned)cntA[i] << 32) | lo;
    unsigned long long b = ((unsigned long long)(unsigned)cntB[i] << 32) | lo;
    if (a > kA) kA = a;
    if (b > kB) kB = b;
  }
  sh[threadIdx.x] = kA; __syncthreads();
  for (int s = 128; s > 0; s >>= 1) {
    if (threadIdx.x < s) {
      unsigned long long o = sh[threadIdx.x + s];
      if (o > sh[threadIdx.x]) sh[threadIdx.x] = o;
    }
    __syncthreads();
  }
  if (threadIdx.x == 0) atomicMax(bestA, sh[0]);
  __syncthreads();
  sh[threadIdx.x] = kB; __syncthreads();
  for (int s = 128; s > 0; s >>= 1) {
    if (threadIdx.x < s) {
      unsigned long long o = sh[threadIdx.x + s];
      if (o > sh[threadIdx.x]) sh[threadIdx.x] = o;
    }
    __syncthreads();
  }
  if (threadIdx.x == 0) atomicMax(bestB, sh[0]);
}

__global__ void k_decode(unsigned long long* bestA, unsigned long long* bestB,
                         int* cntA, int* cntB,
                         int* topLabA, int* topCntA, int* topLabB, int* topCntB,
                         int k) {
  int t = threadIdx.x;
  if (t >= 2) return;
  unsigned long long* best = (t == 0) ? bestA : bestB;
  int* cnts  = (t == 0) ? cntA    : cntB;
  int* tlab  = (t == 0) ? topLabA : topLabB;
  int* tcnt  = (t == 0) ? topCntA : topCntB;
  unsigned long long key = *best;
  unsigned cnt = (unsigned)(key >> 32);
  int lab = (int)(0xFFFFFFFFu - (unsigned)key);
  if (cnt > 0u) {
    tlab[k] = lab; tcnt[k] = (int)cnt; cnts[lab] = 0;
  } else {
    tlab[k] = -1;  tcnt[k] = 0;
  }
  *best = 0ull;  // reset for next pass
}

// ---------------------------------------------------------------------------
// 5) Segment statistics as a 16 x Npix x 16 GEMM on V_WMMA_F32_16X16X4_F32
//    A (16x4): membership of 4 pixels in 16 top segments  (exact 0/1, f32)
//    B (4x16): per-pixel stat columns [f, f^2, aux, y, x, 0...]
//    C (16x16, f32): running sums -> sums[seg][stat]
//
//    Branch-free inner loop: fixed 64-iteration trip count (scalar backedge,
//    EXEC all-ones at the v_wmma per ISA), membership via v_cndmask, column
//    select via per-lane arithmetic weights (pure FMA chain, no control flow).
// ---------------------------------------------------------------------------
__global__ void __launch_bounds__(SEG_TPB)
k_seg_sums(const int* __restrict__ lab, const int* __restrict__ topLab,
           const float* __restrict__ field, const float* __restrict__ aux,
           float* __restrict__ sums /*16x16*/) {
  __shared__ float ls[256];
  for (int i = threadIdx.x; i < 256; i += blockDim.x) ls[i] = 0.0f;
  __syncthreads();

  const int lane = threadIdx.x & 31;
  const int row  = lane & 15;   // A: M index / B: N index
  const int hi   = lane >> 4;   // selects K pair {0,1} vs {2,3}
  const int wave = __builtin_amdgcn_readfirstlane(
      (blockIdx.x * blockDim.x + threadIdx.x) >> 5);
  const int tl   = topLab[row];

  // per-lane column weights (constant across the loop)
  const float w0 = (row == 0) ? 1.0f : 0.0f;
  const float w1 = (row == 1) ? 1.0f : 0.0f;
  const float w2 = (row == 2) ? 1.0f : 0.0f;
  const float w3 = (row == 3) ? 1.0f : 0.0f;
  const float w4 = (row == 4) ? 1.0f : 0.0f;

  const int start = wave * (4 * SEG_CHUNKS) + hi * 2;  // contiguous 256-pixel span

  v8f c = {0.f, 0.f, 0.f, 0.f, 0.f, 0.f, 0.f, 0.f};
#pragma unroll 4
  for (int it = 0; it < SEG_CHUNKS; ++it) {
    const int q0 = start + it * 4;
    const int q1 = q0 + 1;
    const int l0 = lab[q0];
    const int l1 = lab[q1];
    const float f0 = field[q0];
    const float f1 = field[q1];
    const float av0 = aux[q0];
    const float av1 = aux[q1];
    const float y0 = (float)(q0 >> 10), x0 = (float)(q0 & (WW - 1));
    const float y1 = (float)(q1 >> 10), x1 = (float)(q1 & (WW - 1));

    v2f a, b;
    a.x = (l0 == tl) ? 1.0f : 0.0f;
    a.y = (l1 == tl) ? 1.0f : 0.0f;
    // arithmetic column select: FMA chain, branch-free by construction
    b.x = f0 * w0 + (f0 * f0) * w1 + av0 * w2 + y0 * w3 + x0 * w4;
    b.y = f1 * w0 + (f1 * f1) * w1 + av1 * w2 + y1 * w3 + x1 * w4;

    // D = A(16x4) * B(4x16) + C, full f32 accumulate
    c = __builtin_amdgcn_wmma_f32_16x16x4_f32(false, a, false, b,
                                              (short)0, c, false, false);
  }
  // C layout: VGPR r, lanes 0-15 -> M=r, lanes 16-31 -> M=8+r; N = lane&15
#pragma unroll
  for (int r = 0; r < 8; r++)
    atomicAdd(&ls[(r + 8 * hi) * 16 + row], c[r]);
  __syncthreads();
  for (int i = threadIdx.x; i < 256; i += blockDim.x)
    atomicAdd(&sums[i], ls[i]);
}

// ---------------------------------------------------------------------------
// 6) Signatures, centroids, validity, and A->B nearest-neighbor match
// ---------------------------------------------------------------------------
__global__ void k_finalize(const int* __restrict__ topCntA, const int* __restrict__ topCntB,
                           const float* __restrict__ sumsA, const float* __restrict__ sumsB,
                           float* sigA, float* sigB, float* centA, float* centB,
                           float* validA, float* validB, int* match) {
  __shared__ float shSA[KSEG * 4], shSB[KSEG * 4], shVB[KSEG];
  int t = threadIdx.x;
  if (t < 32) {
    int k = t & 15;
    bool isB = (t >= 16);
    const int* tc = isB ? topCntB : topCntA;
    const float* sums = isB ? sumsB : sumsA;
    float* sig  = isB ? sigB  : sigA;
    float* cent = isB ? centB : centA;
    float* val  = isB ? validB : validA;
    int ci = tc[k];
    float cnt = fmaxf((float)ci, 1.0f);
    float mf = sums[k * 16 + 0] / cnt;
    float m2 = sums[k * 16 + 1] / cnt;
    float sf = sqrtf(fmaxf(m2 - mf * mf, 1e-12f));
    float ma = sums[k * 16 + 2] / cnt;
    float s3 = 0.01f * log1pf(cnt);
    sig[k * 4 + 0] = mf; sig[k * 4 + 1] = sf;
    sig[k * 4 + 2] = ma; sig[k * 4 + 3] = s3;
    cent[k * 2 + 0] = sums[k * 16 + 3] / cnt;
    cent[k * 2 + 1] = sums[k * 16 + 4] / cnt;
    float v = (ci >= 20) ? 1.0f : 0.0f;
    val[k] = v;
    float* ss = isB ? shSB : shSA;
    ss[k * 4 + 0] = mf; ss[k * 4 + 1] = sf; ss[k * 4 + 2] = ma; ss[k * 4 + 3] = s3;
    if (isB) shVB[k] = v;
  }
  __syncthreads();
  if (t < 16) {
    float best = 3.4e38f; int bj = 0;
    for (int j = 0; j < KSEG; j++) {
      float d = 0.0f;
      for (int i = 0; i < 4; i++) {
        float e = shSA[t * 4 + i] - shSB[j * 4 + i];
        d += e * e;
      }
      d += (1.0f - shVB[j]) * 1e9f;
      if (d < best) { best = d; bj = j; }
    }
    match[t] = bj;
  }
}

// ---------------------------------------------------------------------------
// 7) Fragment scatter: rotate B segment about its centroid, translate to A
// ---------------------------------------------------------------------------
__global__ void k_scatter(const int* __restrict__ labB, const int* __restrict__ topLabB,
                          const float* __restrict__ sigA, const float* __restrict__ sigB,
                          const float* __restrict__ centA, const float* __restrict__ centB,
                          const float* __restrict__ validA, const float* __restrict__ validB,
                          const int* __restrict__ match, const float* __restrict__ imgB,
                          float* __restrict__ front, float* __restrict__ fcnt,
                          float* __restrict__ scnt) {
  int q = blockIdx.x * blockDim.x + threadIdx.x;
  if (q >= NPIX) return;
  int l = labB[q];
  if (l >= BIGL) return;
  float y = (float)(q >> 10), x = (float)(q & (WW - 1));
  float b0 = imgB[3 * q], b1 = imgB[3 * q + 1], b2 = imgB[3 * q + 2];
  for (int k = 0; k < KSEG; k++) {
    int j = match[k];
    if (topLabB[j] != l) continue;
    float wk = validA[k] * validB[j];
    if (wk <= 0.0f) continue;
    float dy = y - centB[j * 2 + 0];
    float dx = x - centB[j * 2 + 1];
    float ry = dx, rx = -dy;  // 90-degree rotation
    float t0 = 20.0f * tanhf(sigA[k * 4 + 0] - sigB[j * 4 + 0]);
    float t1 = 20.0f * tanhf(sigA[k * 4 + 1] - sigB[j * 4 + 1]);
    float ty = centA[k * 2 + 0] + ry + t0;
    float tx = centA[k * 2 + 1] + rx + t1;
    int iy = (int)fminf(fmaxf(rintf(ty), 0.0f), (float)(HH - 1));
    int ix = (int)fminf(fmaxf(rintf(tx), 0.0f), (float)(WW - 1));
    int flat = iy * WW + ix;
    atomicAdd(&front[3 * flat + 0], b0 * wk);
    atomicAdd(&front[3 * flat + 1], b1 * wk);
    atomicAdd(&front[3 * flat + 2], b2 * wk);
    atomicAdd(&fcnt[flat], wk);
    int sy = (int)fminf(fmaxf(rintf(ty + 7.0f), 0.0f), (float)(HH - 1));
    int sx = (int)fminf(fmaxf(rintf(tx + 7.0f), 0.0f), (float)(WW - 1));
    atomicAdd(&scnt[sy * WW + sx], wk);
  }
}

// ---------------------------------------------------------------------------
// 8) Composite
// ---------------------------------------------------------------------------
__global__ void k_composite(const float* __restrict__ imgA, const float* __restrict__ front,
                            const float* __restrict__ fcnt, const float* __restrict__ scnt,
                            float* __restrict__ out) {
  int p = blockIdx.x * blockDim.x + threadIdx.x;
  if (p >= NPIX) return;
  float fc = fcnt[p], sc = scnt[p];
  float fmask = fminf(fmaxf(fc, 0.0f), 1.0f);
  float smask = fminf(fmaxf(sc, 0.0f), 1.0f);
  float inv = 1.0f / fmaxf(fc, 1.0f);
  float shade = 1.0f - 0.5f * smask;
#pragma unroll
  for (int c = 0; c < 3; c++) {
    float fi = front[3 * p + c] * inv;
    out[3 * p + c] = imgA[3 * p + c] * shade * (1.0f - fmask) + fi * fmask;
  }
}

// ---------------------------------------------------------------------------
// Host orchestration (graph-capturable: only kernels + hipMemsetAsync)
// ---------------------------------------------------------------------------
extern "C" void kernel_launch(void* const* d_in, const int* in_sizes, int n_in,
                              void* d_out, int out_size, void* d_ws, size_t ws_size,
                              hipStream_t stream) {
  const float* imgA  = (const float*)d_in[0];
  const float* imgB  = (const float*)d_in[1];
  const float* fiedA = (const float*)d_in[2];
  const float* fiedB = (const float*)d_in[3];
  const float* gateA = (const float*)d_in[4];
  const unsigned char* contA = (const unsigned char*)d_in[5];
  float* out = (float*)d_out;

  char* ws = (char*)d_ws;
  size_t off = 0;
  auto alloc = [&](size_t bytes) -> char* {
    char* p = ws + off;
    off += (bytes + 255) & ~(size_t)255;
    return p;
  };
  int*   labA0  = (int*)alloc((size_t)NPIX * 4);
  int*   labA1  = (int*)alloc((size_t)NPIX * 4);
  int*   labB0  = (int*)alloc((size_t)NPIX * 4);
  int*   labB1  = (int*)alloc((size_t)NPIX * 4);
  int*   cntA   = (int*)alloc((size_t)NPIX * 4);
  int*   cntB   = (int*)alloc((size_t)NPIX * 4);
  float* front  = (float*)alloc((size_t)NPIX * 3 * 4);
  float* fcnt   = (float*)alloc((size_t)NPIX * 4);
  float* scnt   = (float*)alloc((size_t)NPIX * 4);
  float* auxB   = (float*)alloc((size_t)NPIX * 4);
  unsigned long long* bestA = (unsigned long long*)alloc(8);
  unsigned long long* bestB = (unsigned long long*)alloc(8);
  int*   topLabA = (int*)alloc(KSEG * 4);
  int*   topCntA = (int*)alloc(KSEG * 4);
  int*   topLabB = (int*)alloc(KSEG * 4);
  int*   topCntB = (int*)alloc(KSEG * 4);
  float* sumsA   = (float*)alloc(256 * 4);
  float* sumsB   = (float*)alloc(256 * 4);
  float* sigA    = (float*)alloc(KSEG * 4 * 4);
  float* sigB    = (float*)alloc(KSEG * 4 * 4);
  float* centA   = (float*)alloc(KSEG * 2 * 4);
  float* centB   = (float*)alloc(KSEG * 2 * 4);
  float* validA  = (float*)alloc(KSEG * 4);
  float* validB  = (float*)alloc(KSEG * 4);
  int*   match   = (int*)alloc(KSEG * 4);

  const int TPB = 256;
  const int grid2N = (2 * NPIX + TPB - 1) / TPB;
  const int gridN  = (NPIX + TPB - 1) / TPB;

  // zero scratch that accumulates
  hipMemsetAsync(cntA, 0, (size_t)NPIX * 4, stream);
  hipMemsetAsync(cntB, 0, (size_t)NPIX * 4, stream);
  hipMemsetAsync(front, 0, (size_t)NPIX * 3 * 4, stream);
  hipMemsetAsync(fcnt, 0, (size_t)NPIX * 4, stream);
  hipMemsetAsync(scnt, 0, (size_t)NPIX * 4, stream);
  hipMemsetAsync(bestA, 0, 8, stream);
  hipMemsetAsync(bestB, 0, 8, stream);
  hipMemsetAsync(sumsA, 0, 256 * 4, stream);
  hipMemsetAsync(sumsB, 0, 256 * 4, stream);

  // 1) initial labels + aux plane
  k_init_labels<<<grid2N, TPB, 0, stream>>>(gateA, contA, fiedB, imgB,
                                            labA0, labB0, auxB);

  // 2) 64 CC min-propagation steps (ping-pong; 64 even -> result lands in buf0)
  int* curA = labA0; int* nxtA = labA1;
  int* curB = labB0; int* nxtB = labB1;
  for (int it = 0; it < 64; ++it) {
    k_cc_step<<<grid2N, TPB, 0, stream>>>(curA, nxtA, curB, nxtB);
    int* t;
    t = curA; curA = nxtA; nxtA = t;
    t = curB; curB = nxtB; nxtB = t;
  }

  // 3) bincount
  k_bincount<<<grid2N, TPB, 0, stream>>>(curA, curB, cntA, cntB);

  // 4) top-16 extraction: 16 argmax/decode passes (counts live in L2)
  for (int k = 0; k < KSEG; ++k) {
    k_argmax<<<1024, TPB, 0, stream>>>(cntA, cntB, bestA, bestB);
    k_decode<<<1, 2, 0, stream>>>(bestA, bestB, cntA, cntB,
                                  topLabA, topCntA, topLabB, topCntB, k);
  }

  // 5) segment statistics via WMMA f32 16x16x4 (membership x stats GEMM)
  k_seg_sums<<<SEG_BLOCKS, SEG_TPB, 0, stream>>>(curA, topLabA, fiedA, gateA, sumsA);
  k_seg_sums<<<SEG_BLOCKS, SEG_TPB, 0, stream>>>(curB, topLabB, fiedB, auxB, sumsB);

  // 6) signatures + NN match
  k_finalize<<<1, 32, 0, stream>>>(topCntA, topCntB, sumsA, sumsB,
                                   sigA, sigB, centA, centB, validA, validB, match);

  // 7) fragment scatter
  k_scatter<<<gridN, TPB, 0, stream>>>(curB, topLabB, sigA, sigB, centA, centB,
                                       validA, validB, match, imgB,
                                       front, fcnt, scnt);

  // 8) composite
  k_composite<<<gridN, TPB, 0, stream>>>(imgA, front, fcnt, scnt, out);
}